// PointNet_Pre_SA_AT_Module_55121610277162
// MI455X (gfx1250) — compile-verified
//
#include <hip/hip_runtime.h>
#include <hip/hip_bf16.h>
#include <math.h>

typedef __attribute__((ext_vector_type(2))) float v2f;
typedef __attribute__((ext_vector_type(8))) float v8f;

#define BATCH    8
#define NPTS     4096
#define NPOINT   1024
#define NSAMPLE  32
#define STRIDE   (NPTS / NPOINT)          // 4
#define CIN      6
#define MROWS    (BATCH * NPOINT * NSAMPLE)   // 262144
#define KPAD0    16                           // 9 channels padded to 16
#define BN_EPS   1e-5f

// ---------------------------------------------------------------------------
// Kernel 1: strided sample + kNN (top-32 nearest of 4096) per query.
// One wave (32 lanes) per query; 2 waves / block; distances staged in LDS.
// Also writes new_xyz (first 24576 floats of d_out).
// ---------------------------------------------------------------------------
__global__ void knn_kernel(const float* __restrict__ xyz,
                           int* __restrict__ idx_out,
                           float* __restrict__ out_xyz)
{
    __shared__ float sdist[2][NPTS];          // 32 KB
    const int wave = threadIdx.x >> 5;
    const int lane = threadIdx.x & 31;
    const int q    = blockIdx.x * 2 + wave;   // 0..8191
    const int b    = q / NPOINT;
    const int s    = q % NPOINT;

    const int qbase = (b * NPTS + s * STRIDE) * 3;
    const float qx = xyz[qbase + 0];
    const float qy = xyz[qbase + 1];
    const float qz = xyz[qbase + 2];

    if (lane < 3) out_xyz[q * 3 + lane] = xyz[qbase + lane];

    for (int j = lane; j < NPTS; j += 32) {
        const int pb = (b * NPTS + j) * 3;
        const float dx = qx - xyz[pb + 0];
        const float dy = qy - xyz[pb + 1];
        const float dz = qz - xyz[pb + 2];
        sdist[wave][j] = dx * dx + dy * dy + dz * dz;
    }
    __syncthreads();

    // 32 successive wave-wide argmin extractions (ascending distance order).
    for (int it = 0; it < NSAMPLE; ++it) {
        float mv = __builtin_inff();
        int   mi = 0x7fffffff;
        for (int j = lane; j < NPTS; j += 32) {
            const float v = sdist[wave][j];
            if (v < mv || (v == mv && j < mi)) { mv = v; mi = j; }
        }
        #pragma unroll
        for (int off = 16; off >= 1; off >>= 1) {
            const float ov = __shfl_xor(mv, off, 32);
            const int   oi = __shfl_xor(mi, off, 32);
            if (ov < mv || (ov == mv && oi < mi)) { mv = ov; mi = oi; }
        }
        if (lane == 0) {
            idx_out[q * NSAMPLE + it] = mi;
            sdist[wave][mi] = __builtin_inff();
        }
        __syncthreads();
    }
}

// ---------------------------------------------------------------------------
// Kernel 2: gather neighborhoods -> X0 (MROWS x 16), channels:
//   [0..2] = xyz[idx] - new_xyz,  [3..8] = points[idx],  [9..15] = 0 (pad)
// ---------------------------------------------------------------------------
__global__ void gather_kernel(const float* __restrict__ xyz,
                              const float* __restrict__ points,
                              const int* __restrict__ idx,
                              float* __restrict__ X0)
{
    const int m = blockIdx.x * blockDim.x + threadIdx.x;
    if (m >= MROWS) return;
    const int b   = m / (NPOINT * NSAMPLE);
    const int rem = m % (NPOINT * NSAMPLE);
    const int s   = rem / NSAMPLE;
    const int i   = idx[m];

    const int pb = (b * NPTS + i) * 3;
    const int qb = (b * NPTS + s * STRIDE) * 3;
    float* __restrict__ row = X0 + (size_t)m * KPAD0;
    row[0] = xyz[pb + 0] - xyz[qb + 0];
    row[1] = xyz[pb + 1] - xyz[qb + 1];
    row[2] = xyz[pb + 2] - xyz[qb + 2];
    const float* pp = points + (size_t)(b * NPTS + i) * CIN;
    #pragma unroll
    for (int c = 0; c < CIN; ++c) row[3 + c] = pp[c];
    #pragma unroll
    for (int c = 9; c < KPAD0; ++c) row[c] = 0.0f;
}

// ---------------------------------------------------------------------------
// Kernel 2b: pack W (O x Kreal, row-major) into WMMA B-operand panel order,
// zero-padded to Kpad.  Element e = (((nt*KQ + kq)*32 + lane)*2 + j) holds
// Wt[kq*4 + 2*(lane>>4) + j][nt*16 + (lane&15)], so the GEMM's B fetch is a
// single contiguous global_load_b64 per lane (no bound checks, no exec mask).
// ---------------------------------------------------------------------------
__global__ void pack_w_kernel(const float* __restrict__ W,
                              float* __restrict__ Bpack,
                              int Kpad, int Kreal, int n_tiles)
{
    const int e = blockIdx.x * blockDim.x + threadIdx.x;
    const int KQ = Kpad >> 2;
    const int total = n_tiles * KQ * 64;
    if (e >= total) return;
    const int j    = e & 1;
    const int lane = (e >> 1) & 31;
    const int kq   = (e >> 6) % KQ;
    const int nt   = e / (KQ * 64);
    const int ka   = kq * 4 + 2 * (lane >> 4) + j;
    const int n    = nt * 16 + (lane & 15);
    Bpack[e] = (ka < Kreal) ? W[n * Kreal + ka] : 0.0f;
}

// ---------------------------------------------------------------------------
// Kernel 3: fused [BN+ReLU on input] -> GEMM (Y = act(X) * W^T) via
// V_WMMA_F32_16X16X4_F32.  One wave computes a 16 x (NT*16) strip of Y:
// each A load (with fused activation) feeds NT back-to-back WMMAs.
// K loop fully unrolled (KPAD compile-time), B from the packed panel.
// Per-channel sum / sum-of-squares reduced in-wave then atomically added.
// EXEC is all-ones at every WMMA (grid is an exact multiple of full waves).
//
// VGPR layouts (ISA 7.12.2, 32-bit):
//   A 16x4 : lane = 16*h + r holds {X[m0+r][k0+2h], X[m0+r][k0+2h+1]}
//   B 4x16 : lane = 16*h + c holds {Wt[k0+2h][n0+c], Wt[k0+2h+1][n0+c]}
//   C/D    : VGPR v, lane = 16*h + c  ->  element (m0 + v + 8h, n0 + c)
// ---------------------------------------------------------------------------
template<int KPAD, int NT, bool ACT>
__global__ void gemm_bn_kernel(const float* __restrict__ X,
                               const float* __restrict__ Bpack,
                               float* __restrict__ Y,
                               const float* __restrict__ scale,
                               const float* __restrict__ shift,
                               float* __restrict__ sums,
                               float* __restrict__ sumsq)
{
    constexpr int KQ = KPAD / 4;
    constexpr int O  = NT * 16;

    const int wid  = blockIdx.x * (blockDim.x >> 5) + (threadIdx.x >> 5);
    const int lane = threadIdx.x & 31;
    const int half = lane >> 4;
    const int lidx = lane & 15;
    const int m0   = wid * 16;

    const float* __restrict__ Xrow = X + (size_t)(m0 + lidx) * KPAD;
    const v2f*   __restrict__ Bp   = (const v2f*)Bpack;

    v8f acc[NT];
    #pragma unroll
    for (int t = 0; t < NT; ++t) acc[t] = (v8f){};

    #pragma unroll
    for (int kq = 0; kq < KQ; ++kq) {
        const int ka = kq * 4 + 2 * half;
        float ax = Xrow[ka];
        float ay = Xrow[ka + 1];
        if (ACT) {
            ax = fmaxf(0.0f, ax * scale[ka]     + shift[ka]);
            ay = fmaxf(0.0f, ay * scale[ka + 1] + shift[ka + 1]);
        }
        v2f A; A.x = ax; A.y = ay;
        #pragma unroll
        for (int t = 0; t < NT; ++t) {
            const v2f Bv = Bp[(t * KQ + kq) * 32 + lane];
            acc[t] = __builtin_amdgcn_wmma_f32_16x16x4_f32(
                /*neg_a=*/false, A, /*neg_b=*/false, Bv,
                /*c_mod=*/(short)0, acc[t], /*reuse_a=*/false, /*reuse_b=*/false);
        }
    }

    #pragma unroll
    for (int t = 0; t < NT; ++t) {
        const int n = t * 16 + lidx;
        float csum = 0.0f, csq = 0.0f;
        #pragma unroll
        for (int v = 0; v < 8; ++v) {
            const int row   = m0 + v + 8 * half;
            const float val = acc[t][v];
            Y[(size_t)row * O + n] = val;
            csum += val;
            csq  += val * val;
        }
        csum += __shfl_xor(csum, 16, 32);
        csq  += __shfl_xor(csq,  16, 32);
        if (half == 0) {
            atomicAdd(&sums[n],  csum);
            atomicAdd(&sumsq[n], csq);
        }
    }
}

// ---------------------------------------------------------------------------
// Kernel 4: fold BN stats + gamma/beta into per-channel scale/shift.
// ---------------------------------------------------------------------------
__global__ void bn_stats_kernel(const float* __restrict__ sums,
                                const float* __restrict__ sumsq,
                                const float* __restrict__ gamma,
                                const float* __restrict__ beta,
                                float* __restrict__ scale,
                                float* __restrict__ shift,
                                int O, float invM)
{
    const int o = threadIdx.x;
    if (o >= O) return;
    const float mean = sums[o] * invM;
    const float var  = sumsq[o] * invM - mean * mean;
    const float r    = rsqrtf(var + BN_EPS);
    const float sc   = gamma[o] * r;
    scale[o] = sc;
    shift[o] = beta[o] - mean * sc;
}

// ---------------------------------------------------------------------------
// Kernel 5: BN2 + ReLU fused with max-pool over the 32 neighbors.
// ---------------------------------------------------------------------------
__global__ void finalize_kernel(const float* __restrict__ Y2,
                                const float* __restrict__ scale,
                                const float* __restrict__ shift,
                                float* __restrict__ out_pts)
{
    const int tid = blockIdx.x * blockDim.x + threadIdx.x;  // B*S*128
    if (tid >= BATCH * NPOINT * 128) return;
    const int o  = tid % 128;
    const int bs = tid / 128;
    const float sc = scale[o], sh = shift[o];
    const size_t mbase = (size_t)bs * NSAMPLE;
    float mx = 0.0f;                       // relu output >= 0
    #pragma unroll 4
    for (int k = 0; k < NSAMPLE; ++k) {
        const float v = fmaxf(0.0f, Y2[(mbase + k) * 128 + o] * sc + sh);
        mx = fmaxf(mx, v);
    }
    out_pts[(size_t)bs * 128 + o] = mx;
}

// ---------------------------------------------------------------------------
extern "C" void kernel_launch(void* const* d_in, const int* in_sizes, int n_in,
                              void* d_out, int out_size, void* d_ws, size_t ws_size,
                              hipStream_t stream)
{
    const float* xyz    = (const float*)d_in[0];
    const float* points = (const float*)d_in[1];
    const float* W0 = (const float*)d_in[2];
    const float* g0 = (const float*)d_in[3];
    const float* b0 = (const float*)d_in[4];
    const float* W1 = (const float*)d_in[5];
    const float* g1 = (const float*)d_in[6];
    const float* b1 = (const float*)d_in[7];
    const float* W2 = (const float*)d_in[8];
    const float* g2 = (const float*)d_in[9];
    const float* b2 = (const float*)d_in[10];

    float* out_xyz = (float*)d_out;                              // 8*1024*3
    float* out_pts = out_xyz + BATCH * NPOINT * 3;               // 8*1024*128

    // ---- workspace carve-up ----
    char* ws = (char*)d_ws;
    int*   idx   = (int*)ws;                       ws += (size_t)MROWS * sizeof(int);     // 1 MB
    float* stats = (float*)ws;                     ws += 1024 * sizeof(float);            // 4 KB
    float* sums0 = stats,        *sq0 = stats + 64;
    float* scl0  = stats + 128,  *shf0 = stats + 192;
    float* sums1 = stats + 256,  *sq1 = stats + 320;
    float* scl1  = stats + 384,  *shf1 = stats + 448;
    float* sums2 = stats + 512,  *sq2 = stats + 640;
    float* scl2  = stats + 768,  *shf2 = stats + 896;
    float* Bp0 = (float*)ws;  ws += 4 * 4  * 64 * sizeof(float);  // 4 KB
    float* Bp1 = (float*)ws;  ws += 4 * 16 * 64 * sizeof(float);  // 16 KB
    float* Bp2 = (float*)ws;  ws += 8 * 16 * 64 * sizeof(float);  // 32 KB
    float* X0 = (float*)ws;  ws += (size_t)MROWS * KPAD0 * sizeof(float);  // 16 MB
    float* Y0 = (float*)ws;  ws += (size_t)MROWS * 64    * sizeof(float);  // 64 MB
    float* Y1 = (float*)ws;  ws += (size_t)MROWS * 64    * sizeof(float);  // 64 MB
    float* Y2 = (float*)ws;                                                // 128 MB

    const float invM = 1.0f / (float)MROWS;

    // zero BN accumulators each call (graph-replay deterministic; async = capture-legal)
    hipMemsetAsync(stats, 0, 1024 * sizeof(float), stream);

    // 1) sampling + kNN (writes new_xyz into d_out)
    knn_kernel<<<BATCH * NPOINT / 2, 64, 0, stream>>>(xyz, idx, out_xyz);

    // 2) gather grouped features + pack weight panels
    gather_kernel<<<MROWS / 256, 256, 0, stream>>>(xyz, points, idx, X0);
    pack_w_kernel<<<(4 * 4  * 64 + 255) / 256, 256, 0, stream>>>(W0, Bp0, 16, 9,  4);
    pack_w_kernel<<<(4 * 16 * 64 + 255) / 256, 256, 0, stream>>>(W1, Bp1, 64, 64, 4);
    pack_w_kernel<<<(8 * 16 * 64 + 255) / 256, 256, 0, stream>>>(W2, Bp2, 64, 64, 8);

    const int gemm_blocks = (MROWS / 16) / 8;   // one wave per 16-row strip

    // 3) layer 0 : (M x 16[9]) @ W0^T -> Y0 (M x 64), collect stats
    gemm_bn_kernel<16, 4, false><<<gemm_blocks, 256, 0, stream>>>(
        X0, Bp0, Y0, nullptr, nullptr, sums0, sq0);
    bn_stats_kernel<<<1, 64, 0, stream>>>(sums0, sq0, g0, b0, scl0, shf0, 64, invM);

    // 4) layer 1 : act0(Y0) @ W1^T -> Y1 (M x 64)
    gemm_bn_kernel<64, 4, true><<<gemm_blocks, 256, 0, stream>>>(
        Y0, Bp1, Y1, scl0, shf0, sums1, sq1);
    bn_stats_kernel<<<1, 64, 0, stream>>>(sums1, sq1, g1, b1, scl1, shf1, 64, invM);

    // 5) layer 2 : act1(Y1) @ W2^T -> Y2 (M x 128)
    gemm_bn_kernel<64, 8, true><<<gemm_blocks, 256, 0, stream>>>(
        Y1, Bp2, Y2, scl1, shf1, sums2, sq2);
    bn_stats_kernel<<<1, 128, 0, stream>>>(sums2, sq2, g2, b2, scl2, shf2, 128, invM);

    // 6) BN2 + ReLU + max over neighbors -> new_points
    finalize_kernel<<<(BATCH * NPOINT * 128) / 256, 256, 0, stream>>>(Y2, scl2, shf2, out_pts);
}